// GCNEncoder_29635274342568
// MI455X (gfx1250) — compile-verified
//
#include <hip/hip_runtime.h>

// ============================================================================
// GCN encoder (VGAE) for MI455X / gfx1250.
//
// Perf model: edge aggregation (~218M fp32 L2 atomics; per-layer h <= 25.6MB
// fits in the 192MB L2) dominates; dense GEMMs are ~2.25 GFLOP total -> use
// exact-fp32 WMMA (V_WMMA_F32_16X16X4_F32), precision is free at this
// arithmetic intensity. GEMM epilogue fuses the dinv row-scaling and the
// self-loop init, eliminating one full N*F pass per layer.
// ============================================================================

typedef __attribute__((ext_vector_type(2))) float v2f;
typedef __attribute__((ext_vector_type(8))) float v8f;

// ---------------------------------------------------------------------------
// Degree / dinv
// ---------------------------------------------------------------------------
__global__ void deg_init_kernel(float* __restrict__ deg, int n) {
  int i = blockIdx.x * blockDim.x + threadIdx.x;
  if (i < n) deg[i] = 1.0f;  // self-loop contribution
}

__global__ void deg_accum_kernel(float* __restrict__ deg,
                                 const long long* __restrict__ dst,
                                 long long e_cnt) {
  long long e = (long long)blockIdx.x * blockDim.x + threadIdx.x;
  if (e < e_cnt) unsafeAtomicAdd(&deg[dst[e]], 1.0f);  // native global_atomic_add_f32
}

__global__ void dinv_kernel(float* __restrict__ deg, int n) {
  int i = blockIdx.x * blockDim.x + threadIdx.x;
  if (i < n) deg[i] = __frsqrt_rn(deg[i]);  // deg >= 1 always
}

// ---------------------------------------------------------------------------
// Fused GEMM + row scale:  h = X[M,K] * W[K,N];  G = O = dinv[row] * h.
// One wave per 16-row M tile (EXEC all-ones for WMMA); NT = N/16 col tiles.
// M must be a multiple of 16 (100000 = 6250*16).
//
// fp32 WMMA lane layouts (ISA 7.12.2):
//   A 16x4 : lane L holds row M=L%16; VGPR0/1 = K = (L<16 ? 0,1 : 2,3)
//   B 4x16 : lane L holds col N=L%16; VGPR0/1 = K = (L<16 ? 0,1 : 2,3)
//   C 16x16: lane L holds col N=L%16; VGPR v  = row M = v + (L<16 ? 0 : 8)
//
// W is staged in LDS K-pair-packed: pair row kp holds (K=2kp, K=2kp+1)
// interleaved per column, so a B fragment is ONE aligned ds_load_b64.
// Pair-row stride is forced to ===32 dwords (mod 64 banks) so the two
// half-waves (which read adjacent kp rows) hit disjoint bank halves.
// ---------------------------------------------------------------------------
template <int K, int NT>
__global__ void gcn_gemm_scale(const float* __restrict__ X,
                               const float* __restrict__ W,
                               const float* __restrict__ dinv,
                               float* __restrict__ G,   // g = dinv*h (scatter src)
                               float* __restrict__ O,   // accumulator init (self-loop)
                               int MT) {
  constexpr int N      = NT * 16;
  constexpr int STRIDE = ((2 * N) % 64 == 32) ? (2 * N) : (2 * N + 32);  // dwords
  __shared__ alignas(16) float sW[(K / 2) * STRIDE];

  for (int i = threadIdx.x; i < K * N; i += blockDim.x) {
    int k = i / N, n = i % N;
    sW[(k >> 1) * STRIDE + n * 2 + (k & 1)] = W[i];
  }
  __syncthreads();

  const int lane = threadIdx.x & 31;
  const int mt   = blockIdx.x * (blockDim.x >> 5) + (threadIdx.x >> 5);
  if (mt >= MT) return;  // whole-wave exit

  const int ncol   = lane & 15;        // N (and A-row) index within tile
  const int khalf  = (lane >> 4) << 1; // 0 | 2
  const int kp_off = lane >> 4;        // 0 | 1 : kp row offset for this half-wave
  const float* __restrict__ xrow = X + (size_t)(mt * 16 + ncol) * K;

  v8f acc[NT] = {};
#pragma unroll 4
  for (int k0 = 0; k0 < K; k0 += 4) {
    v2f a = *(const v2f*)(xrow + k0 + khalf);  // 8B-aligned (k0+khalf even)
#pragma unroll
    for (int t = 0; t < NT; ++t) {
      v2f b = *(const v2f*)&sW[(k0 / 2 + kp_off) * STRIDE + (t * 16 + ncol) * 2];
      acc[t] = __builtin_amdgcn_wmma_f32_16x16x4_f32(
          /*neg_a=*/false, a, /*neg_b=*/false, b,
          /*c_mod=*/(short)0, acc[t], /*reuse_a=*/false, /*reuse_b=*/false);
    }
  }

  // Epilogue: scale rows by dinv and write g + self-loop-initialized out.
  const int mbase = mt * 16 + ((lane >> 4) << 3);
  float dv[8];
#pragma unroll
  for (int v = 0; v < 8; ++v) dv[v] = dinv[mbase + v];
#pragma unroll
  for (int t = 0; t < NT; ++t)
#pragma unroll
    for (int v = 0; v < 8; ++v) {
      float val = acc[t][v] * dv[v];
      size_t o  = (size_t)(mbase + v) * N + t * 16 + ncol;
      G[o] = val;
      O[o] = val;
    }
}

// ---------------------------------------------------------------------------
// out[dst] += g[src]; thread = (edge, 4 features): float4 gather + 4 atomics.
// Atomic count (E*F) is the roofline limiter; everything else is minimized.
// ---------------------------------------------------------------------------
template <int F>
__global__ void edge_scatter_kernel(const float* __restrict__ g,
                                    const long long* __restrict__ src,
                                    const long long* __restrict__ dst,
                                    float* __restrict__ out,
                                    long long e_cnt) {
  constexpr int FQ = F / 4;
  long long idx = (long long)blockIdx.x * blockDim.x + threadIdx.x;
  if (idx >= e_cnt * FQ) return;
  long long e = idx / FQ;
  int       f = (int)(idx % FQ) * 4;
  long long s = src[e];
  long long d = dst[e];
  float4 gv = *(const float4*)(g + s * F + f);  // 16B-aligned: f%4==0, F%16==0
  float* op = out + d * F + f;
  unsafeAtomicAdd(op + 0, gv.x);
  unsafeAtomicAdd(op + 1, gv.y);
  unsafeAtomicAdd(op + 2, gv.z);
  unsafeAtomicAdd(op + 3, gv.w);
}

// ---------------------------------------------------------------------------
// out = act(dinv[i] * out + bias[f])   (F power of two -> shift/mask)
// ---------------------------------------------------------------------------
template <int F>
__global__ void finalize_kernel(float* __restrict__ out,
                                const float* __restrict__ dinv,
                                const float* __restrict__ bias,
                                int n, int relu) {
  int idx = blockIdx.x * blockDim.x + threadIdx.x;
  if (idx >= n * F) return;
  float v = dinv[idx / F] * out[idx] + bias[idx % F];
  out[idx] = relu ? fmaxf(v, 0.0f) : v;
}

// ---------------------------------------------------------------------------
// Orchestration
// ---------------------------------------------------------------------------
extern "C" void kernel_launch(void* const* d_in, const int* in_sizes, int n_in,
                              void* d_out, int out_size, void* d_ws, size_t ws_size,
                              hipStream_t stream) {
  const float*     x   = (const float*)d_in[0];
  const long long* ei  = (const long long*)d_in[1];
  const float*     W1  = (const float*)d_in[2];
  const float*     b1  = (const float*)d_in[3];
  const float*     W2  = (const float*)d_in[4];
  const float*     b2  = (const float*)d_in[5];
  const float*     Wmu = (const float*)d_in[6];
  const float*     bmu = (const float*)d_in[7];
  const float*     Wls = (const float*)d_in[8];
  const float*     bls = (const float*)d_in[9];

  const int       N = in_sizes[0] / 128;          // 100000 (multiple of 16)
  const long long E = (long long)in_sizes[1] / 2; // 1600000
  const long long* src = ei;
  const long long* dst = ei + E;

  // Workspace: dinv[N] | bufA[N*64] | bufB[N*64]  (~52 MB)
  float* dinv = (float*)d_ws;
  float* bufA = dinv + N;
  float* bufB = bufA + (size_t)N * 64;
  float* mu   = (float*)d_out;                    // [N,16]
  float* ls   = mu + (size_t)N * 16;              // [N,16]

  const int T = 256;
  auto cdiv = [](long long a, long long b) { return (int)((a + b - 1) / b); };
  const int MT = N / 16;            // 6250 wave-tiles
  const int gB = cdiv(MT, T / 32);  // 8 waves per block

  // --- normalization coefficients ---
  deg_init_kernel <<<cdiv(N, T), T, 0, stream>>>(dinv, N);
  deg_accum_kernel<<<cdiv(E, T), T, 0, stream>>>(dinv, dst, E);
  dinv_kernel     <<<cdiv(N, T), T, 0, stream>>>(dinv, N);

  // --- layer 1: 128 -> 64, relu ---
  gcn_gemm_scale<128, 4><<<gB, T, 0, stream>>>(x, W1, dinv, bufA, bufB, MT);
  edge_scatter_kernel<64><<<cdiv(E * 16, T), T, 0, stream>>>(bufA, src, dst, bufB, E);
  finalize_kernel<64><<<cdiv((long long)N * 64, T), T, 0, stream>>>(bufB, dinv, b1, N, 1);
  // x1 = bufB

  // --- layer 2: 64 -> 32, relu ---
  float* h2 = bufA;                   // g2 [N,32]
  float* o2 = bufA + (size_t)N * 32;  // [N,32] -> becomes x2
  gcn_gemm_scale<64, 2><<<gB, T, 0, stream>>>(bufB, W2, dinv, h2, o2, MT);
  edge_scatter_kernel<32><<<cdiv(E * 8, T), T, 0, stream>>>(h2, src, dst, o2, E);
  finalize_kernel<32><<<cdiv((long long)N * 32, T), T, 0, stream>>>(o2, dinv, b2, N, 1);
  // x2 = o2

  // --- heads: 32 -> 16 (mu, logstd), no activation, aggregate in d_out ---
  float* hmu = bufB;                  // reuse (x1 dead)
  float* hls = bufB + (size_t)N * 16;
  gcn_gemm_scale<32, 1><<<gB, T, 0, stream>>>(o2, Wmu, dinv, hmu, mu, MT);
  gcn_gemm_scale<32, 1><<<gB, T, 0, stream>>>(o2, Wls, dinv, hls, ls, MT);
  edge_scatter_kernel<16><<<cdiv(E * 4, T), T, 0, stream>>>(hmu, src, dst, mu, E);
  edge_scatter_kernel<16><<<cdiv(E * 4, T), T, 0, stream>>>(hls, src, dst, ls, E);
  finalize_kernel<16><<<cdiv((long long)N * 16, T), T, 0, stream>>>(mu, dinv, bmu, N, 0);
  finalize_kernel<16><<<cdiv((long long)N * 16, T), T, 0, stream>>>(ls, dinv, bls, N, 0);
}